// RobotGraphClassify_60979945669188
// MI455X (gfx1250) — compile-verified
//
#include <hip/hip_runtime.h>
#include <math.h>

#define NODES 50000
#define EDGES 1600000
#define NT (NODES / 16)   // 3125 row tiles of 16 nodes

typedef __attribute__((ext_vector_type(2))) float v2f;
typedef __attribute__((ext_vector_type(8))) float v8f;

__device__ __forceinline__ float lrelu(float v) { return v >= 0.0f ? v : 0.01f * v; }

__device__ __forceinline__ v8f wmma4(v2f a, v2f b, v8f c) {
    // D = A(16x4 f32) x B(4x16 f32) + C(16x16 f32)
    return __builtin_amdgcn_wmma_f32_16x16x4_f32(false, a, false, b, (short)0, c, false, false);
}

// ---------------------------------------------------------------------------
// Kernel 1: fused feature encoder -> x1[N,64]
//   cols  0..15 : leaky(screen @ Ws + bs)     (768->16, WMMA)
//   cols 16..31 : leaky(des    @ Wd + bd)     (768->16, WMMA)
//   cols 32..47 : leaky(tweet  @ Wt + bt)     (768->16, WMMA)
//   cols 48..55 : leaky(profile @ Wp + bp)    (5->8, VALU)
//   cols 56..63 : leaky(personal @ Wpe + bpe) (7->8, VALU)
// One wave per 16-node tile.
// ---------------------------------------------------------------------------
__global__ void feat_kernel(const float* __restrict__ screen, const float* __restrict__ des,
                            const float* __restrict__ tweet, const float* __restrict__ profile,
                            const float* __restrict__ personal,
                            const float* __restrict__ Ws, const float* __restrict__ bs,
                            const float* __restrict__ Wd, const float* __restrict__ bd,
                            const float* __restrict__ Wt, const float* __restrict__ bt,
                            const float* __restrict__ Wp, const float* __restrict__ bp,
                            const float* __restrict__ Wpe, const float* __restrict__ bpe,
                            float* __restrict__ x1) {
    int gw = (int)((blockIdx.x * blockDim.x + threadIdx.x) >> 5);  // wave-uniform
    if (gw >= NT) return;
    int lane = threadIdx.x & 31;
    int m = lane & 15;
    int koff = (lane >> 4) << 1;
    int row0 = gw << 4;

    const float* Xs[3] = {screen, des, tweet};
    const float* Wb[3] = {Ws, Wd, Wt};
    const float* Bb[3] = {bs, bd, bt};

#pragma unroll
    for (int g = 0; g < 3; ++g) {
        const float* __restrict__ X = Xs[g];
        const float* __restrict__ W = Wb[g];
        const float* __restrict__ arow = X + (size_t)(row0 + m) * 768;
        v8f c = {};
        for (int k = 0; k < 768; k += 4) {
            v2f a;
            a.x = arow[k + koff];
            a.y = arow[k + koff + 1];
            v2f b;
            b.x = W[(size_t)(k + koff) * 16 + m];
            b.y = W[(size_t)(k + koff + 1) * 16 + m];
            c = wmma4(a, b, c);
        }
        int colBase = g * 16;
        float bv = Bb[g][m];
        int rb = row0 + ((lane >> 4) << 3);
#pragma unroll
        for (int i = 0; i < 8; ++i)
            x1[(size_t)(rb + i) * 64 + colBase + m] = lrelu(c[i] + bv);
    }

    // small branches: lanes 0..15 -> profile (K=5), lanes 16..31 -> personal (K=7)
    {
        int half = lane >> 4;
        const float* Xp = half ? personal : profile;
        const float* Wsm = half ? Wpe : Wp;
        const float* bsm = half ? bpe : bp;
        int K = half ? 7 : 5;
        int colBase = 48 + half * 8;
        const float* xr = Xp + (size_t)(row0 + m) * K;
        float xin[7];
        for (int k = 0; k < 7; ++k) xin[k] = (k < K) ? xr[k] : 0.0f;
#pragma unroll
        for (int o = 0; o < 8; ++o) {
            float acc = bsm[o];
            for (int k = 0; k < K; ++k) acc += xin[k] * Wsm[k * 8 + o];
            x1[(size_t)(row0 + m) * 64 + colBase + o] = lrelu(acc);
        }
    }
}

// ---------------------------------------------------------------------------
// Generic Y = X[N,64] @ W[64,64] (+ bias, + leaky if mode==1). Wave per tile.
// ---------------------------------------------------------------------------
__global__ void gemm64_kernel(const float* __restrict__ X, const float* __restrict__ W,
                              const float* __restrict__ bias, float* __restrict__ Y, int mode) {
    int gw = (int)((blockIdx.x * blockDim.x + threadIdx.x) >> 5);
    if (gw >= NT) return;
    int lane = threadIdx.x & 31;
    int m = lane & 15;
    int koff = (lane >> 4) << 1;
    int row0 = gw << 4;
    const float* __restrict__ arow = X + (size_t)(row0 + m) * 64;

    v8f c0 = {}, c1 = {}, c2 = {}, c3 = {};
    for (int k = 0; k < 64; k += 4) {
        v2f a;
        a.x = arow[k + koff];
        a.y = arow[k + koff + 1];
        const float* w0 = W + (size_t)(k + koff) * 64;
        const float* w1 = w0 + 64;
        v2f b;
        b.x = w0[m];      b.y = w1[m];      c0 = wmma4(a, b, c0);
        b.x = w0[16 + m]; b.y = w1[16 + m]; c1 = wmma4(a, b, c1);
        b.x = w0[32 + m]; b.y = w1[32 + m]; c2 = wmma4(a, b, c2);
        b.x = w0[48 + m]; b.y = w1[48 + m]; c3 = wmma4(a, b, c3);
    }

    int rb = row0 + ((lane >> 4) << 3);
    v8f cc[4] = {c0, c1, c2, c3};
#pragma unroll
    for (int t = 0; t < 4; ++t) {
        int col = t * 16 + m;
        float bv = mode ? bias[col] : 0.0f;
#pragma unroll
        for (int i = 0; i < 8; ++i) {
            float v = cc[t][i] + bv;
            if (mode) v = lrelu(v);
            Y[(size_t)(rb + i) * 64 + col] = v;
        }
    }
}

// ---------------------------------------------------------------------------
// Degree / normalization (shared by all 3 GCN layers)
// ---------------------------------------------------------------------------
__global__ void deg_init_kernel(float* __restrict__ deg) {
    int i = blockIdx.x * blockDim.x + threadIdx.x;
    if (i < NODES) deg[i] = 1.0f;  // self-loop contribution
}
__global__ void deg_count_kernel(float* __restrict__ deg, const int* __restrict__ col) {
    int i = blockIdx.x * blockDim.x + threadIdx.x;
    if (i < EDGES) atomicAdd(&deg[col[i]], 1.0f);
}
__global__ void dinv_kernel(float* __restrict__ deg) {
    int i = blockIdx.x * blockDim.x + threadIdx.x;
    if (i < NODES) {
        float d = deg[i];
        deg[i] = d > 0.0f ? rsqrtf(d) : 0.0f;
    }
}

// Y[i,j] = bias[j] + x1[i,j] + h[i,j]*dinv[i]^2   (bias + residual + self-loop)
__global__ void gcn_init_kernel(float* __restrict__ Y, const float* __restrict__ H,
                                const float* __restrict__ X1, const float* __restrict__ dinv,
                                const float* __restrict__ bias) {
    size_t i = (size_t)blockIdx.x * blockDim.x + threadIdx.x;
    if (i >= (size_t)NODES * 64) return;
    int node = (int)(i >> 6);
    int j = (int)(i & 63);
    float di = dinv[node];
    Y[i] = bias[j] + X1[i] + H[i] * di * di;
}

// Wave per edge: Y[col] += H[row] * dinv[row]*dinv[col]  (64 floats, 2 per lane)
__global__ void scatter_kernel(float* __restrict__ Y, const float* __restrict__ H,
                               const int* __restrict__ row, const int* __restrict__ col,
                               const float* __restrict__ dinv) {
    int gw = (int)((blockIdx.x * blockDim.x + threadIdx.x) >> 5);
    if (gw >= EDGES) return;
    int lane = threadIdx.x & 31;
    int r = row[gw];
    int c = col[gw];
    float nrm = dinv[r] * dinv[c];
    const float* __restrict__ hr = H + (size_t)r * 64;
    float* __restrict__ yc = Y + (size_t)c * 64;
    atomicAdd(&yc[lane], hr[lane] * nrm);
    atomicAdd(&yc[lane + 32], hr[lane + 32] * nrm);
}

// ---------------------------------------------------------------------------
// Head: y = leaky(x @ W_o1 + b_o1) [WMMA] -> LDS -> out = sigmoid(y @ W_o2 + b_o2)
// 4 waves / block, 16KB LDS.
// ---------------------------------------------------------------------------
__global__ void head_kernel(const float* __restrict__ X, const float* __restrict__ W1,
                            const float* __restrict__ b1, const float* __restrict__ W2,
                            const float* __restrict__ b2, float* __restrict__ out) {
    __shared__ float ylds[4 * 16 * 64];
    int wib = threadIdx.x >> 5;
    int gw = (int)((blockIdx.x * blockDim.x + threadIdx.x) >> 5);
    int lane = threadIdx.x & 31;
    int m = lane & 15;
    int koff = (lane >> 4) << 1;
    int row0 = gw << 4;
    bool active = (gw < NT);  // wave-uniform

    if (active) {
        const float* __restrict__ arow = X + (size_t)(row0 + m) * 64;
        v8f c0 = {}, c1 = {}, c2 = {}, c3 = {};
        for (int k = 0; k < 64; k += 4) {
            v2f a;
            a.x = arow[k + koff];
            a.y = arow[k + koff + 1];
            const float* w0 = W1 + (size_t)(k + koff) * 64;
            const float* w1 = w0 + 64;
            v2f b;
            b.x = w0[m];      b.y = w1[m];      c0 = wmma4(a, b, c0);
            b.x = w0[16 + m]; b.y = w1[16 + m]; c1 = wmma4(a, b, c1);
            b.x = w0[32 + m]; b.y = w1[32 + m]; c2 = wmma4(a, b, c2);
            b.x = w0[48 + m]; b.y = w1[48 + m]; c3 = wmma4(a, b, c3);
        }
        float* yb = ylds + wib * 16 * 64;
        int rloc = (lane >> 4) << 3;
        v8f cc[4] = {c0, c1, c2, c3};
#pragma unroll
        for (int t = 0; t < 4; ++t) {
            int col = t * 16 + m;
            float bv = b1[col];
#pragma unroll
            for (int i = 0; i < 8; ++i) yb[(rloc + i) * 64 + col] = lrelu(cc[t][i] + bv);
        }
    }
    __syncthreads();
    if (active) {
        const float* yr = ylds + wib * 16 * 64 + (lane >> 1) * 64;
        int o = lane & 1;
        float acc = b2[o];
        for (int k = 0; k < 64; ++k) acc += yr[k] * W2[k * 2 + o];
        out[(size_t)(row0 + (lane >> 1)) * 2 + o] = 1.0f / (1.0f + expf(-acc));
    }
}

// ---------------------------------------------------------------------------
extern "C" void kernel_launch(void* const* d_in, const int* in_sizes, int n_in,
                              void* d_out, int out_size, void* d_ws, size_t ws_size,
                              hipStream_t stream) {
    (void)in_sizes; (void)n_in; (void)out_size; (void)ws_size;

    const float* screen   = (const float*)d_in[0];
    const float* des      = (const float*)d_in[1];
    const float* tweet    = (const float*)d_in[2];
    const float* profile  = (const float*)d_in[3];
    const float* personal = (const float*)d_in[4];
    const int*   edge     = (const int*)d_in[5];
    const int*   erow     = edge;
    const int*   ecol     = edge + EDGES;
    const float* W_screen = (const float*)d_in[6];  const float* b_screen = (const float*)d_in[7];
    const float* W_des    = (const float*)d_in[8];  const float* b_des    = (const float*)d_in[9];
    const float* W_tweet  = (const float*)d_in[10]; const float* b_tweet  = (const float*)d_in[11];
    const float* W_pro    = (const float*)d_in[12]; const float* b_pro    = (const float*)d_in[13];
    const float* W_per    = (const float*)d_in[14]; const float* b_per    = (const float*)d_in[15];
    const float* W_lin    = (const float*)d_in[16]; const float* b_lin    = (const float*)d_in[17];
    const float* W_g1     = (const float*)d_in[18]; const float* b_g1     = (const float*)d_in[19];
    const float* W_g2     = (const float*)d_in[20]; const float* b_g2     = (const float*)d_in[21];
    const float* W_g3     = (const float*)d_in[22]; const float* b_g3     = (const float*)d_in[23];
    const float* W_o1     = (const float*)d_in[24]; const float* b_o1     = (const float*)d_in[25];
    const float* W_o2     = (const float*)d_in[26]; const float* b_o2     = (const float*)d_in[27];

    const size_t NF = (size_t)NODES * 64;
    float* ws   = (float*)d_ws;
    float* x1   = ws;            // [N,64]
    float* xA   = x1 + NF;       // [N,64]
    float* xB   = xA + NF;       // [N,64]
    float* h    = xB + NF;       // [N,64]
    float* dinv = h + NF;        // [N]  (deg -> dinv in place)

    dim3 b256(256);
    int tileGrid = (NT + 7) / 8;  // 8 waves per 256-thread block

    // 1) fused feature encoder -> x1
    feat_kernel<<<tileGrid, b256, 0, stream>>>(screen, des, tweet, profile, personal,
                                               W_screen, b_screen, W_des, b_des, W_tweet, b_tweet,
                                               W_pro, b_pro, W_per, b_per, x1);

    // 2) x = leaky(x1 @ W_lin + b_lin)
    gemm64_kernel<<<tileGrid, b256, 0, stream>>>(x1, W_lin, b_lin, xA, 1);

    // 3) degree normalization (once, shared by all 3 GCN layers)
    deg_init_kernel<<<(NODES + 255) / 256, b256, 0, stream>>>(dinv);
    deg_count_kernel<<<(EDGES + 255) / 256, b256, 0, stream>>>(dinv, ecol);
    dinv_kernel<<<(NODES + 255) / 256, b256, 0, stream>>>(dinv);

    // 4) three GCN layers with x1 residual
    const float* Wg[3] = {W_g1, W_g2, W_g3};
    const float* bg[3] = {b_g1, b_g2, b_g3};
    float* xin = xA;
    float* xout = xB;
    for (int l = 0; l < 3; ++l) {
        gemm64_kernel<<<tileGrid, b256, 0, stream>>>(xin, Wg[l], nullptr, h, 0);
        gcn_init_kernel<<<(int)((NF + 255) / 256), b256, 0, stream>>>(xout, h, x1, dinv, bg[l]);
        scatter_kernel<<<(EDGES + 7) / 8, b256, 0, stream>>>(xout, h, erow, ecol, dinv);
        float* t = xin; xin = xout; xout = t;
    }

    // 5) head: leaky(x @ W_o1 + b_o1) -> sigmoid(y @ W_o2 + b_o2)
    head_kernel<<<(NT + 3) / 4, dim3(128), 0, stream>>>(xin, W_o1, b_o1, W_o2, b_o2, (float*)d_out);
}